// KNearestNeighbor_31559419691267
// MI455X (gfx1250) — compile-verified
//
#include <hip/hip_runtime.h>

typedef float v2f __attribute__((ext_vector_type(2)));
typedef float v8f __attribute__((ext_vector_type(8)));

#define BATCH   4
#define DIM     64
#define NPTS    8192
#define KNN     8
#define QTILE   16
#define NWAVES  4
#define RSTRIPE (NPTS / NWAVES)   // 2048 ref points per wave
#define RTILES  (RSTRIPE / 16)    // 128 tiles of 16 ref points

__launch_bounds__(128, 1)
__global__ void knn_wmma_kernel(const float* __restrict__ ref,
                                const float* __restrict__ query,
                                int* __restrict__ out)
{
    __shared__ float s_rn[NPTS];                    // 32 KB: ||ref_r||^2 for whole batch slice
    __shared__ float s_cd[QTILE][NWAVES * 2][KNN];  // 4 KB candidate distances
    __shared__ int   s_ci[QTILE][NWAVES * 2][KNN];  // 4 KB candidate indices

    const int tid  = threadIdx.x;
    const int wave = tid >> 5;
    const int lane = tid & 31;
    const int m    = lane & 15;   // column within 16-wide tile
    const int h    = lane >> 4;   // half-wave select (K-pair / row-half select)

    const int b     = blockIdx.x / (NPTS / QTILE);
    const int qtile = (blockIdx.x % (NPTS / QTILE)) * QTILE;

    const float* refb = ref   + (size_t)b * DIM * NPTS;
    const float* qryb = query + (size_t)b * DIM * NPTS;

    // ---------------- rn[r] = sum_d ref[d,r]^2 into LDS (coalesced) ----------
    for (int i = 0; i < NPTS / 128; ++i) {
        const int r = tid + 128 * i;
        const float* p = refb + r;
        float acc = 0.0f;
        #pragma unroll 8
        for (int d = 0; d < DIM; ++d) {
            float v = p[(size_t)d * NPTS];
            acc = fmaf(v, v, acc);
        }
        s_rn[r] = acc;
    }
    __syncthreads();

    // ---------------- preload query tile as WMMA B operand (32 VGPRs) -------
    // B chunk kk: lane holds B[k = 2h + {0,1} (+4kk), n = m] = query[d, qtile+m]
    v2f qB[16];
    {
        const float* qp = qryb + (size_t)(2 * h) * NPTS + qtile + m;
        #pragma unroll
        for (int kk = 0; kk < 16; ++kk) {
            v2f v;
            v.x = qp[(size_t)(4 * kk) * NPTS];
            v.y = qp[(size_t)(4 * kk) * NPTS + NPTS];
            qB[kk] = v;
        }
    }

    // ---------------- per-lane sorted top-8 (ascending) ----------------------
    float hd[KNN];
    int   hi[KNN];
    #pragma unroll
    for (int k = 0; k < KNN; ++k) { hd[k] = 3.0e38f; hi[k] = 0; }

    const int rstripe = wave * RSTRIPE;
    const float* apb = refb + (size_t)(2 * h) * NPTS + m;

    for (int t = 0; t < RTILES; ++t) {
        const int rbase = rstripe + t * 16;

        // A chunk kk: lane holds A[mrow = m, k = 2h + {0,1} (+4kk)] = ref[d, rbase+m]
        const float* ap = apb + rbase;
        v2f aA[16];
        #pragma unroll
        for (int kk = 0; kk < 16; ++kk) {
            v2f v;
            v.x = ap[(size_t)(4 * kk) * NPTS];
            v.y = ap[(size_t)(4 * kk) * NPTS + NPTS];
            aA[kk] = v;
        }

        // cross^T tile: acc[g] (lane L) = sum_d ref[d, rbase+8h+g] * query[d, qtile+m]
        v8f acc = {};
        #pragma unroll
        for (int kk = 0; kk < 16; ++kk) {
            acc = __builtin_amdgcn_wmma_f32_16x16x4_f32(
                false, aA[kk], false, qB[kk], (short)0, acc, false, false);
        }

        // score s = rn[r] - 2*cross  (qn[q] is a per-query constant: irrelevant to top-k)
        #pragma unroll
        for (int g = 0; g < 8; ++g) {
            const int r = rbase + 8 * h + g;
            const float s = fmaf(-2.0f, acc[g], s_rn[r]);
            const bool pred = s < hd[KNN - 1];
            if (__builtin_amdgcn_ballot_w32(pred)) {   // wave-uniform skip
                // conditionally replace worst, then one bubble pass restores order
                hd[KNN - 1] = pred ? s : hd[KNN - 1];
                hi[KNN - 1] = pred ? r : hi[KNN - 1];
                #pragma unroll
                for (int k = KNN - 1; k > 0; --k) {
                    const float a = hd[k - 1], c = hd[k];
                    const int  ai = hi[k - 1], ci = hi[k];
                    const bool sw = c < a;
                    hd[k - 1] = sw ? c : a;   hd[k] = sw ? a : c;
                    hi[k - 1] = sw ? ci : ai; hi[k] = sw ? ai : ci;
                }
            }
        }
    }

    // ---------------- dump per-lane lists, merge 8 lists per query ----------
    {
        const int slot = wave * 2 + h;   // lane owns query qtile+m, candidate slot
        #pragma unroll
        for (int k = 0; k < KNN; ++k) {
            s_cd[m][slot][k] = hd[k];
            s_ci[m][slot][k] = hi[k];
        }
    }
    __syncthreads();

    if (tid < QTILE) {
        const int q = tid;
        float bd[KNN];
        int   bi[KNN];
        #pragma unroll
        for (int k = 0; k < KNN; ++k) { bd[k] = 3.0e38f; bi[k] = 0; }

        for (int slot = 0; slot < NWAVES * 2; ++slot) {
            #pragma unroll
            for (int k = 0; k < KNN; ++k) {
                const float c  = s_cd[q][slot][k];
                const int   ci = s_ci[q][slot][k];
                if (c >= bd[KNN - 1]) break;   // source list sorted: rest can't help
                bd[KNN - 1] = c;
                bi[KNN - 1] = ci;
                #pragma unroll
                for (int j = KNN - 1; j > 0; --j) {
                    const float a = bd[j - 1], d = bd[j];
                    const int  ai = bi[j - 1], di = bi[j];
                    const bool sw = d < a;
                    bd[j - 1] = sw ? d : a;   bd[j] = sw ? a : d;
                    bi[j - 1] = sw ? di : ai; bi[j] = sw ? ai : di;
                }
            }
        }

        // out[b][k][qtile+q] = 1-based ref index
        const size_t obase = (size_t)b * KNN * NPTS + (qtile + q);
        #pragma unroll
        for (int k = 0; k < KNN; ++k)
            out[obase + (size_t)k * NPTS] = bi[k] + 1;
    }
}

extern "C" void kernel_launch(void* const* d_in, const int* in_sizes, int n_in,
                              void* d_out, int out_size, void* d_ws, size_t ws_size,
                              hipStream_t stream)
{
    const float* ref   = (const float*)d_in[0];   // [B, D, N] f32
    const float* query = (const float*)d_in[1];   // [B, D, N] f32
    int* out = (int*)d_out;                       // [B, K, N] int32 (1-based)

    (void)in_sizes; (void)n_in; (void)out_size; (void)d_ws; (void)ws_size;

    const int blocks = BATCH * (NPTS / QTILE);    // 4 * 512 = 2048 workgroups
    knn_wmma_kernel<<<dim3(blocks), dim3(128), 0, stream>>>(ref, query, out);
}